// RobustInverseCompositional_22462678958152
// MI455X (gfx1250) — compile-verified
//
#include <hip/hip_runtime.h>
#include <math.h>

#define BATCH    8
#define IMH      512
#define IMW      512
#define NPIX     (IMH * IMW)          // 262144 per batch image
#define DELTA    10
#define LAM2     25.0f
#define TOLF     1e-3f
#define MAX_ITER 15

#define NBLK     256                  // solver blocks (32 per batch) -- safely co-resident
#define BLKPB    (NBLK / BATCH)       // 32 blocks per batch
#define ROWSPB   (IMH / BLKPB)        // 16 rows per block
#define NTHR     256                  // 8 waves (wave32)

// ---- ws layout (bytes) ----
// [0]        : unsigned arrive counter (reset each launch by init kernel)
// [64..127]  : float p_warp[16] (written by solver, read by finalize)
// [256..]    : double partials[2][BATCH][BLKPB][5]   (double-buffered by iter parity)
#define WS_PWARP_OFF   64
#define WS_PART_OFF    256

typedef float v2f __attribute__((ext_vector_type(2)));
typedef float v8f __attribute__((ext_vector_type(8)));

// Sum a value across all 32 lanes of a wave.
// CDNA5 path: one V_WMMA_F32_16X16X4_F32 with B = ones. With A VGPR0 = v and
// A VGPR1 = 0, row-sum(m) = v[m] + v[m+16]; D register j holds row-sum(j) in
// lanes 0-15 and row-sum(j+8) in lanes 16-31, so Sum_j D[j] gives the half-wave
// sums and one shfl_xor(16) completes the full 32-lane sum (all lanes).
__device__ __forceinline__ float waveSum32(float v) {
#if __has_builtin(__builtin_amdgcn_wmma_f32_16x16x4_f32)
    v2f a;    a[0] = v;    a[1] = 0.0f;
    v2f ones; ones[0] = 1.0f; ones[1] = 1.0f;
    v8f c = {0.f, 0.f, 0.f, 0.f, 0.f, 0.f, 0.f, 0.f};
    v8f d = __builtin_amdgcn_wmma_f32_16x16x4_f32(
        /*neg_a=*/false, a, /*neg_b=*/false, ones,
        /*c_mod=*/(short)0, c, /*reuse_a=*/false, /*reuse_b=*/false);
    float s = ((d[0] + d[1]) + (d[2] + d[3])) + ((d[4] + d[5]) + (d[6] + d[7]));
    s += __shfl_xor(s, 16, 32);
    return s;
#else
    #pragma unroll
    for (int off = 16; off > 0; off >>= 1) v += __shfl_xor(v, off, 32);
    return v;
#endif
}

// Fast reciprocal sqrt: single v_rsq_f32 (~1 ulp) instead of IEEE sqrt + divide.
__device__ __forceinline__ float fastRsq(float x) {
#if __has_builtin(__builtin_amdgcn_rsqf)
    return __builtin_amdgcn_rsqf(x);
#else
    return 1.0f / sqrtf(x);
#endif
}

__global__ void ric_init_kernel(unsigned* counter) {
    if (threadIdx.x == 0) *counter = 0u;
}

__global__ void __launch_bounds__(NTHR)
ric_solver_kernel(const float* __restrict__ I1, const float* __restrict__ I2,
                  const float* __restrict__ p_in, float* __restrict__ out,
                  unsigned char* __restrict__ ws)
{
    unsigned* counter = (unsigned*)ws;
    float*    pwarp_g = (float*)(ws + WS_PWARP_OFF);
    double*   partials = (double*)(ws + WS_PART_OFF);

    const int tid  = threadIdx.x;
    const int b    = blockIdx.x / BLKPB;   // batch handled by this block
    const int blk  = blockIdx.x % BLKPB;   // block index within batch
    const int yBase = blk * ROWSPB;
    const float* __restrict__ I1b = I1 + (size_t)b * NPIX;
    const float* __restrict__ I2b = I2 + (size_t)b * NPIX;

    __shared__ float s_pAll[16];   // full p state, tracked identically in every block
    __shared__ float s_pWarp[16];  // p of the most recent ACTIVE warp
    __shared__ float s_err;
    __shared__ float sp[2];        // this block's batch translation, broadcast
    __shared__ float sred[NTHR / 32][5];

    if (tid == 0) {
        #pragma unroll
        for (int k = 0; k < 16; ++k) { float v = p_in[k]; s_pAll[k] = v; s_pWarp[k] = v; }
        s_err = 1e10f;
        sp[0] = s_pAll[2 * b]; sp[1] = s_pAll[2 * b + 1];
    }
    __syncthreads();

    const int warp = tid >> 5;

    for (int it = 0; it < MAX_ITER; ++it) {
        const float dx = sp[0];
        const float dy = sp[1];
        float hxx = 0.f, hxy = 0.f, hyy = 0.f, bxs = 0.f, bys = 0.f;

        for (int r = 0; r < ROWSPB; ++r) {
            const int y = yBase + r;
            if (y < DELTA || y >= IMH - DELTA) continue;   // G==0 rows contribute nothing
            const float yw  = (float)y + dy;
            const float fy0 = fminf(fmaxf(floorf(yw), 0.f), (float)(IMH - 2));
            const int   y0  = (int)fy0;
            const float wy  = yw - fy0;
            const bool  yValid = (yw >= 0.f) && (yw <= (float)(IMH - 1));
            const float* __restrict__ rowC = I1b + y * IMW;
            const float* __restrict__ rowU = I1b + (y - 1) * IMW;
            const float* __restrict__ rowD = I1b + (y + 1) * IMW;
            const float* __restrict__ w0   = I2b + y0 * IMW;
            const float* __restrict__ w1   = w0 + IMW;
            // keep next rows warm in the WGP cache (global_prefetch_b8)
            __builtin_prefetch(rowD + IMW + tid, 0, 3);
            __builtin_prefetch(w1 + IMW + tid, 0, 3);
            #pragma unroll
            for (int h = 0; h < 2; ++h) {
                const int x = tid + h * NTHR;
                if (x < DELTA || x >= IMW - DELTA) continue;  // G==0 cols contribute nothing
                const float Gx  = 0.5f * (rowC[x + 1] - rowC[x - 1]);
                const float Gy  = 0.5f * (rowD[x] - rowU[x]);
                const float xw  = (float)x + dx;
                const float fx0 = fminf(fmaxf(floorf(xw), 0.f), (float)(IMW - 2));
                const int   x0  = (int)fx0;
                const float wx  = xw - fx0;
                const bool  valid = yValid && (xw >= 0.f) && (xw <= (float)(IMW - 1));
                const float I00 = w0[x0], I01 = w0[x0 + 1];
                const float I10 = w1[x0], I11 = w1[x0 + 1];
                const float Iw  = (1.f - wy) * ((1.f - wx) * I00 + wx * I01)
                                +        wy  * ((1.f - wx) * I10 + wx * I11);
                const float di  = valid ? (Iw - rowC[x]) : 0.f;  // DI_filled
                const float rho = fastRsq(di * di + LAM2);       // rho covers invalid too
                hxx += rho * Gx * Gx;  hxy += rho * Gx * Gy;  hyy += rho * Gy * Gy;
                bxs += rho * Gx * di;  bys += rho * Gy * di;
            }
        }

        // ---- wave reduction via WMMA (EXEC is full here: all threads participate) ----
        const float s0 = waveSum32(hxx);
        const float s1 = waveSum32(hxy);
        const float s2 = waveSum32(hyy);
        const float s3 = waveSum32(bxs);
        const float s4 = waveSum32(bys);
        if ((tid & 31) == 0) {
            sred[warp][0] = s0; sred[warp][1] = s1; sred[warp][2] = s2;
            sred[warp][3] = s3; sred[warp][4] = s4;
        }
        __syncthreads();

        if (tid == 0) {
            double t0 = 0, t1 = 0, t2 = 0, t3 = 0, t4 = 0;
            #pragma unroll
            for (int w2 = 0; w2 < NTHR / 32; ++w2) {
                t0 += (double)sred[w2][0]; t1 += (double)sred[w2][1];
                t2 += (double)sred[w2][2]; t3 += (double)sred[w2][3];
                t4 += (double)sred[w2][4];
            }
            double* slot = partials + ((((size_t)(it & 1) * BATCH + b) * BLKPB + blk) * 5);
            slot[0] = t0; slot[1] = t1; slot[2] = t2; slot[3] = t3; slot[4] = t4;
        }

        // ---- grid barrier (release -> arrive -> spin -> acquire) ----
        __syncthreads();
        if (tid == 0) {
            __threadfence();
            atomicAdd(counter, 1u);
            const unsigned target = (unsigned)(NBLK * (it + 1));
            while (__hip_atomic_load(counter, __ATOMIC_RELAXED, __HIP_MEMORY_SCOPE_AGENT) < target) {
                __builtin_amdgcn_s_sleep(2);
            }
            __threadfence();
        }
        __syncthreads();

        // ---- redundant (identical) 2x2 solve on every block's thread 0 ----
        if (tid == 0) {
            const double* base = partials + (size_t)(it & 1) * BATCH * BLKPB * 5;
            float dp[16];
            double errsq = 0.0;
            #pragma unroll
            for (int bb = 0; bb < BATCH; ++bb) {
                double S0 = 0, S1 = 0, S2 = 0, S3 = 0, S4 = 0;
                for (int k = 0; k < BLKPB; ++k) {
                    const double* q = base + ((size_t)bb * BLKPB + k) * 5;
                    S0 += q[0]; S1 += q[1]; S2 += q[2]; S3 += q[3]; S4 += q[4];
                }
                const double det = S0 * S2 - S1 * S1;
                const double inv = 1.0 / det;
                const float dpx = (float)((S2 * S3 - S1 * S4) * inv);
                const float dpy = (float)((S0 * S4 - S1 * S3) * inv);
                dp[2 * bb] = dpx; dp[2 * bb + 1] = dpy;
                errsq += (double)dpx * dpx + (double)dpy * dpy;
            }
            const float errN = sqrtf((float)errsq);
            if (s_err > TOLF) {            // gate uses PREVIOUS err, like the reference
                #pragma unroll
                for (int k = 0; k < 16; ++k) { s_pWarp[k] = s_pAll[k]; s_pAll[k] -= dp[k]; }
                s_err = errN;
            }
            sp[0] = s_pAll[2 * b]; sp[1] = s_pAll[2 * b + 1];
        }
        __syncthreads();
    }

    if (blockIdx.x == 0 && tid == 0) {
        #pragma unroll
        for (int k = 0; k < 16; ++k) { out[k] = s_pAll[k]; pwarp_g[k] = s_pWarp[k]; }
        out[16] = s_err;
    }
}

__global__ void __launch_bounds__(256)
ric_finalize_kernel(const float* __restrict__ I1, const float* __restrict__ I2,
                    float* __restrict__ out, const unsigned char* __restrict__ ws)
{
    const float* pwarp = (const float*)(ws + WS_PWARP_OFF);
    float* __restrict__ DIo = out + 17;
    float* __restrict__ IWo = out + 17 + (size_t)BATCH * NPIX;
    const int gid    = blockIdx.x * 256 + threadIdx.x;  // 2048 blocks * 256 = 524288
    const int stride = 2048 * 256;
    const float nanv = __int_as_float(0x7FC00000);
    #pragma unroll
    for (int k = 0; k < 4; ++k) {
        const int pix = gid + k * stride;               // < 2097152
        const int b   = pix / NPIX;
        const int rem = pix - b * NPIX;
        const int y   = rem >> 9;
        const int x   = rem & (IMW - 1);
        const float dx = pwarp[2 * b], dy = pwarp[2 * b + 1];
        const float xw = (float)x + dx, yw = (float)y + dy;
        const bool valid = (xw >= 0.f) && (xw <= (float)(IMW - 1)) &&
                           (yw >= 0.f) && (yw <= (float)(IMH - 1));
        const float fx0 = fminf(fmaxf(floorf(xw), 0.f), (float)(IMW - 2));
        const float fy0 = fminf(fmaxf(floorf(yw), 0.f), (float)(IMH - 2));
        const int x0 = (int)fx0, y0 = (int)fy0;
        const float wx = xw - fx0, wy = yw - fy0;
        const float* __restrict__ r0 = I2 + (size_t)b * NPIX + y0 * IMW;
        const float I00 = r0[x0], I01 = r0[x0 + 1];
        const float I10 = r0[x0 + IMW], I11 = r0[x0 + IMW + 1];
        const float Iw = (1.f - wy) * ((1.f - wx) * I00 + wx * I01)
                       +        wy  * ((1.f - wx) * I10 + wx * I11);
        const float di = Iw - I1[(size_t)b * NPIX + rem];
        DIo[pix] = valid ? di : nanv;
        IWo[pix] = valid ? Iw : nanv;
    }
}

extern "C" void kernel_launch(void* const* d_in, const int* in_sizes, int n_in,
                              void* d_out, int out_size, void* d_ws, size_t ws_size,
                              hipStream_t stream)
{
    (void)in_sizes; (void)n_in; (void)out_size; (void)ws_size;
    const float* I1 = (const float*)d_in[0];
    const float* I2 = (const float*)d_in[1];
    const float* p  = (const float*)d_in[2];
    float* out = (float*)d_out;
    unsigned char* ws = (unsigned char*)d_ws;

    ric_init_kernel<<<1, 32, 0, stream>>>((unsigned*)ws);
    ric_solver_kernel<<<NBLK, NTHR, 0, stream>>>(I1, I2, p, out, ws);
    ric_finalize_kernel<<<2048, 256, 0, stream>>>(I1, I2, out, ws);
}